// NGCF_27719718928490
// MI455X (gfx1250) — compile-verified
//
#include <hip/hip_runtime.h>

#define N_USERS 100000
#define N_ITEMS 100000
#define N_NODES (N_USERS + N_ITEMS)
#define NNZ     3200000
#define EMB     64
#define HOPS    3
#define ASTRIDE 68   // padded LDS row stride (floats) to spread banks

typedef __attribute__((ext_vector_type(2))) float v2f;
typedef __attribute__((ext_vector_type(8))) float v8f;

// ---------------------------------------------------------------------------
// init: ego = concat(user_emb, item_emb); out[:,0,:] = ego
// ---------------------------------------------------------------------------
__global__ __launch_bounds__(256) void ngcf_init_kernel(
    const float* __restrict__ user, const float* __restrict__ item,
    float* __restrict__ ego, float* __restrict__ out)
{
    size_t idx4 = (size_t)blockIdx.x * blockDim.x + threadIdx.x;
    if (idx4 >= (size_t)N_NODES * (EMB / 4)) return;
    size_t flat = idx4 * 4;
    size_t node = flat >> 6;
    int d = (int)(flat & 63);
    const float* src = (node < N_USERS) ? (user + node * EMB + d)
                                        : (item + (node - N_USERS) * EMB + d);
    float4 v = *(const float4*)src;
    *(float4*)(ego + flat) = v;
    *(float4*)(out + node * ((HOPS + 1) * EMB) + d) = v;   // hop-0 slice
}

// ---------------------------------------------------------------------------
// zero: side = 0
// ---------------------------------------------------------------------------
__global__ __launch_bounds__(256) void ngcf_zero_kernel(float* __restrict__ p, int n)
{
    int i = (blockIdx.x * blockDim.x + threadIdx.x) * 4;
    if (i < n) *(float4*)(p + i) = make_float4(0.f, 0.f, 0.f, 0.f);
}

// ---------------------------------------------------------------------------
// scatter: side[rows[e]] += vals[e] * ego[cols[e]]   (one wave32 per edge)
// lane l handles dims {2l, 2l+1}; native f32 atomics via inline asm
// ---------------------------------------------------------------------------
__global__ __launch_bounds__(256) void ngcf_scatter_kernel(
    const int* __restrict__ rows, const int* __restrict__ cols,
    const float* __restrict__ vals, const float* __restrict__ ego,
    float* __restrict__ side)
{
    const int edge = blockIdx.x * 8 + (threadIdx.x >> 5);
    const int lane = threadIdx.x & 31;
    if (edge >= NNZ) return;
    const int r = __builtin_amdgcn_readfirstlane(rows[edge]);
    const int c = __builtin_amdgcn_readfirstlane(cols[edge]);
    const float v = vals[edge];
    const float2 e = *(const float2*)(ego + (size_t)c * EMB + lane * 2);
    const float m0 = e.x * v;
    const float m1 = e.y * v;
    unsigned long long addr =
        (unsigned long long)(side + (size_t)r * EMB + lane * 2);
    asm volatile("global_atomic_add_f32 %0, %1, off"
                 :: "v"(addr), "v"(m0) : "memory");
    asm volatile("global_atomic_add_f32 %0, %1, off"
                 :: "v"(addr + 4ull), "v"(m1) : "memory");
}

// ---------------------------------------------------------------------------
// per-hop dense stage (WMMA f32 16x16x4):
//   sum = side @ Wgc + bgc ; bi = (ego*side) @ Wbi + bbi
//   egoNext = leaky_relu(sum+bi, 0.2) ; outSlice = normalize(egoNext, dim=1)
// block = 64 threads = 2 waves; each wave owns 16 node rows.
// ---------------------------------------------------------------------------
__global__ __launch_bounds__(64) void ngcf_gemm_kernel(
    const float* __restrict__ side, const float* __restrict__ ego,
    const float* __restrict__ Wgc,  const float* __restrict__ bgc,
    const float* __restrict__ Wbi,  const float* __restrict__ bbi,
    float* __restrict__ egoNext,    float* __restrict__ outSlice)
{
    __shared__ float sWgc[EMB * EMB];
    __shared__ float sWbi[EMB * EMB];
    __shared__ float sBias[EMB];
    __shared__ float sSide[2][16 * ASTRIDE];
    __shared__ float sProd[2][16 * ASTRIDE];

    const int tid  = threadIdx.x;
    const int w    = tid >> 5;
    const int lane = tid & 31;
    const int nodeBase = blockIdx.x * 32 + w * 16;

    // stage W matrices (4096 floats each) + combined bias
    for (int i = tid * 4; i < EMB * EMB; i += 64 * 4) {
        *(float4*)&sWgc[i] = *(const float4*)&Wgc[i];
        *(float4*)&sWbi[i] = *(const float4*)&Wbi[i];
    }
    sBias[tid] = bgc[tid] + bbi[tid];

    // stage this wave's 16x64 side tile and (ego*side) tile (coalesced float4)
    for (int p = 0; p < 8; ++p) {
        int flat = p * 128 + lane * 4;
        int row = flat >> 6, col = flat & 63;
        size_t g = (size_t)(nodeBase + row) * EMB + col;
        float4 s4 = *(const float4*)&side[g];
        float4 e4 = *(const float4*)&ego[g];
        *(float4*)&sSide[w][row * ASTRIDE + col] = s4;
        float4 p4 = make_float4(s4.x * e4.x, s4.y * e4.y, s4.z * e4.z, s4.w * e4.w);
        *(float4*)&sProd[w][row * ASTRIDE + col] = p4;
    }
    __syncthreads();

    // A 16x4 f32 fragment layout: lanes 0-15 -> M=lane, K={k0,k0+1};
    //                             lanes 16-31 -> M=lane-16, K={k0+2,k0+3}
    const int lrow  = lane & 15;
    const int khalf = (lane < 16) ? 0 : 2;
    const int rowOff = (lane < 16) ? 0 : 8;   // C/D: VGPR r holds M=r / M=r+8

    float fout[4][8];
    float nrm2[8];
#pragma unroll
    for (int r = 0; r < 8; ++r) nrm2[r] = 0.f;

#pragma unroll
    for (int nt = 0; nt < 4; ++nt) {
        v8f accGC = {};
        v8f accBI = {};
        const int ncol = nt * 16 + lrow;
#pragma unroll
        for (int k0 = 0; k0 < EMB; k0 += 4) {
            const int kk = k0 + khalf;
            v2f aS = *(const v2f*)&sSide[w][lrow * ASTRIDE + kk];
            v2f aP = *(const v2f*)&sProd[w][lrow * ASTRIDE + kk];
            v2f bG, bB;
            bG.x = sWgc[kk * EMB + ncol];
            bG.y = sWgc[(kk + 1) * EMB + ncol];
            bB.x = sWbi[kk * EMB + ncol];
            bB.y = sWbi[(kk + 1) * EMB + ncol];
            accGC = __builtin_amdgcn_wmma_f32_16x16x4_f32(
                false, aS, false, bG, (short)0, accGC, false, false);
            accBI = __builtin_amdgcn_wmma_f32_16x16x4_f32(
                false, aP, false, bB, (short)0, accBI, false, false);
        }
        const float bias = sBias[ncol];
#pragma unroll
        for (int r = 0; r < 8; ++r) {
            float f = accGC[r] + accBI[r] + bias;
            f = f > 0.f ? f : 0.2f * f;            // leaky_relu(0.2)
            fout[nt][r] = f;
            // next-hop ego is the UN-normalized activation
            egoNext[(size_t)(nodeBase + rowOff + r) * EMB + ncol] = f;
            // row sum of squares: reduce across the 16-lane half
            float s2 = f * f;
            s2 += __shfl_xor(s2, 1);
            s2 += __shfl_xor(s2, 2);
            s2 += __shfl_xor(s2, 4);
            s2 += __shfl_xor(s2, 8);
            nrm2[r] += s2;
        }
    }

#pragma unroll
    for (int r = 0; r < 8; ++r) {
        float inv = 1.0f / fmaxf(sqrtf(nrm2[r]), 1e-12f);
#pragma unroll
        for (int nt = 0; nt < 4; ++nt) {
            outSlice[(size_t)(nodeBase + rowOff + r) * ((HOPS + 1) * EMB)
                     + nt * 16 + lrow] = fout[nt][r] * inv;
        }
    }
}

// ---------------------------------------------------------------------------
extern "C" void kernel_launch(void* const* d_in, const int* in_sizes, int n_in,
                              void* d_out, int out_size, void* d_ws, size_t ws_size,
                              hipStream_t stream)
{
    const int*   rows = (const int*)d_in[0];
    const int*   cols = (const int*)d_in[1];
    const float* vals = (const float*)d_in[2];
    const float* user = (const float*)d_in[3];
    const float* item = (const float*)d_in[4];
    const float* Wgc  = (const float*)d_in[5];
    const float* bgc  = (const float*)d_in[6];
    const float* Wbi  = (const float*)d_in[7];
    const float* bbi  = (const float*)d_in[8];
    float* out = (float*)d_out;

    const size_t nodeFloats = (size_t)N_NODES * EMB;   // 12.8M floats
    float* ego_a = (float*)d_ws;
    float* ego_b = ego_a + nodeFloats;
    float* sideB = ego_b + nodeFloats;

    const int vec4Blocks = (int)((nodeFloats / 4 + 255) / 256);   // 12500

    ngcf_init_kernel<<<vec4Blocks, 256, 0, stream>>>(user, item, ego_a, out);

    float* cur = ego_a;
    float* nxt = ego_b;
    for (int k = 0; k < HOPS; ++k) {
        ngcf_zero_kernel<<<vec4Blocks, 256, 0, stream>>>(sideB, (int)nodeFloats);
        ngcf_scatter_kernel<<<(NNZ + 7) / 8, 256, 0, stream>>>(
            rows, cols, vals, cur, sideB);
        ngcf_gemm_kernel<<<N_NODES / 32, 64, 0, stream>>>(
            sideB, cur,
            Wgc + (size_t)k * EMB * EMB, bgc + (size_t)k * EMB,
            Wbi + (size_t)k * EMB * EMB, bbi + (size_t)k * EMB,
            nxt, out + (size_t)(k + 1) * EMB);
        float* t = cur; cur = nxt; nxt = t;
    }
}